// ASAScorer_14044543057997
// MI455X (gfx1250) — compile-verified
//
#include <hip/hip_runtime.h>
#include <hip/hip_bf16.h>

#define DCH 64
#define NEG_SLOPE 0.2f
#define SM_EPS 1e-16f
#define ENC_NEG_INF 0x007FFFFFu   // enc(-inf)

typedef __attribute__((ext_vector_type(2))) float v2f;
typedef __attribute__((ext_vector_type(8))) float v8f;

// ---- order-preserving float <-> uint encoding for atomicMax-based segment max ----
__device__ __forceinline__ unsigned enc_f(float f) {
    unsigned u = __float_as_uint(f);
    return (u & 0x80000000u) ? ~u : (u | 0x80000000u);
}
__device__ __forceinline__ float dec_f(unsigned e) {
    unsigned u = (e & 0x80000000u) ? (e & 0x7FFFFFFFu) : ~e;
    return __uint_as_float(u);
}
__device__ __forceinline__ float wave_sum32(float v) {
#pragma unroll
    for (int off = 16; off > 0; off >>= 1) v += __shfl_xor(v, off, 32);
    return v;
}

// ---------------- K1: init accumulators ----------------
__global__ void k_init(unsigned* __restrict__ xq_enc, unsigned* __restrict__ m_enc,
                       float* __restrict__ denom, float* __restrict__ agg, int n) {
    int i = blockIdx.x * blockDim.x + threadIdx.x;
    if (i < n * DCH) xq_enc[i] = ENC_NEG_INF;
    if (i < n) {
        m_enc[i] = ENC_NEG_INF;
        denom[i] = 0.0f;
        agg[i]   = 0.0f;
    }
}

// ---------------- K2: x_q = segment_max(x[row], col) ----------------
// one thread per (edge, channel); lanes of a wave share an edge row -> coalesced
__global__ void k_segmax(const float* __restrict__ x, const int* __restrict__ row,
                         const int* __restrict__ col, unsigned* __restrict__ xq_enc, int E) {
    int tid = blockIdx.x * blockDim.x + threadIdx.x;
    int e = tid >> 6, d = tid & 63;
    if (e >= E) return;
    unsigned v = enc_f(x[row[e] * DCH + d]);
    atomicMax(&xq_enc[col[e] * DCH + d], v);
}

// ---------------- K3: x_q_lin = x_q @ lin_w.T + lin_b  (WMMA f32 16x16x4) ----------------
// One wave per 16-row tile, K=64 in 16 steps, 4 column tiles of 16.
// A frag (16x4 f32, ISA 7.12.2): lane m=lane&15; lane<16 holds K={k0,k0+1}, lane>=16 K={k0+2,k0+3}.
// B frag (4x16) assumed mirrored: lane n=lane&15, same K split.
// C/D (16x16 f32): vgpr v, lane<16 -> M=v, lane>=16 -> M=v+8; N=lane&15.
__global__ void k_gemm_wmma(const unsigned* __restrict__ xq_enc, const float* __restrict__ lin_w,
                            const float* __restrict__ lin_b, float* __restrict__ xqlin, int n) {
    int wave = (blockIdx.x * blockDim.x + threadIdx.x) >> 5;
    int lane = threadIdx.x & 31;
    int tiles = (n + 15) >> 4;
    if (wave >= tiles) return;            // wave-uniform exit: EXEC all-1s for WMMA
    int row0 = wave << 4;
    int r = row0 + (lane & 15);
    int rc = r < n ? r : n - 1;           // clamp reads for (unused) tail tile
    int koff = (lane < 16) ? 0 : 2;
    v8f acc[4] = {};
#pragma unroll
    for (int k0 = 0; k0 < DCH; k0 += 4) {
        v2f a;
        a.x = dec_f(xq_enc[rc * DCH + k0 + koff]);
        a.y = dec_f(xq_enc[rc * DCH + k0 + koff + 1]);
#pragma unroll
        for (int nt = 0; nt < 4; ++nt) {
            int ncol = nt * 16 + (lane & 15);
            v2f b;
            b.x = lin_w[ncol * DCH + k0 + koff];       // lin_w.T[k][ncol] = lin_w[ncol][k]
            b.y = lin_w[ncol * DCH + k0 + koff + 1];
            acc[nt] = __builtin_amdgcn_wmma_f32_16x16x4_f32(
                false, a, false, b, (short)0, acc[nt], false, false);
        }
    }
    int mbase = (lane < 16) ? 0 : 8;
#pragma unroll
    for (int nt = 0; nt < 4; ++nt) {
        int ncol = nt * 16 + (lane & 15);
        float bias = lin_b[ncol];
#pragma unroll
        for (int v = 0; v < 8; ++v) {
            int orow = row0 + mbase + v;
            if (orow < n) xqlin[orow * DCH + ncol] = acc[nt][v] + bias;
        }
    }
}

// ---------------- K4: per-node attention dots; zero xp (aliases xqlin buffer) ----------------
__global__ void k_node_scores(float* __restrict__ buf /* xqlin in, xp(=0) out */,
                              const float* __restrict__ x, const float* __restrict__ att_w,
                              const float* __restrict__ att_b, float* __restrict__ sA,
                              float* __restrict__ sB, int n) {
    int node = (blockIdx.x * blockDim.x + threadIdx.x) >> 5;
    int lane = threadIdx.x & 31;
    if (node >= n) return;
    float q0 = buf[node * DCH + lane];
    float q1 = buf[node * DCH + lane + 32];
    float x0 = x[node * DCH + lane];
    float x1 = x[node * DCH + lane + 32];
    float sa = wave_sum32(q0 * att_w[lane] + q1 * att_w[lane + 32]);
    float sb = wave_sum32(x0 * att_w[64 + lane] + x1 * att_w[96 + lane]);
    if (lane == 0) { sA[node] = sa + att_b[0]; sB[node] = sb; }
    // buffer reuse: zero xp after reads (per-thread same-address ordering)
    buf[node * DCH + lane] = 0.0f;
    buf[node * DCH + lane + 32] = 0.0f;
}

__device__ __forceinline__ float edge_score(const float* sA, const float* sB, int r, int c) {
    float s = sA[c] + sB[r];
    return (s >= 0.0f) ? s : NEG_SLOPE * s;
}

// ---------------- K5: segment softmax max ----------------
__global__ void k_edge_max(const int* __restrict__ row, const int* __restrict__ col,
                           const float* __restrict__ sA, const float* __restrict__ sB,
                           unsigned* __restrict__ m_enc, int E) {
    int e = blockIdx.x * blockDim.x + threadIdx.x;
    if (e >= E) return;
    int r = row[e], c = col[e];
    atomicMax(&m_enc[c], enc_f(edge_score(sA, sB, r, c)));
}

// ---------------- K6: segment softmax denominator ----------------
__global__ void k_edge_denom(const int* __restrict__ row, const int* __restrict__ col,
                             const float* __restrict__ sA, const float* __restrict__ sB,
                             const unsigned* __restrict__ m_enc, float* __restrict__ denom, int E) {
    int e = blockIdx.x * blockDim.x + threadIdx.x;
    if (e >= E) return;
    int r = row[e], c = col[e];
    float ex = __expf(edge_score(sA, sB, r, c) - dec_f(m_enc[c]));
    atomicAdd(&denom[c], ex);
}

// ---------------- K7: xp = segment_sum(x[row] * softmax_weight, col) ----------------
__global__ void k_edge_scatter(const float* __restrict__ x, const int* __restrict__ row,
                               const int* __restrict__ col, const float* __restrict__ sA,
                               const float* __restrict__ sB, const unsigned* __restrict__ m_enc,
                               const float* __restrict__ denom, float* __restrict__ xp, int E) {
    int tid = blockIdx.x * blockDim.x + threadIdx.x;
    int e = tid >> 6, d = tid & 63;
    if (e >= E) return;
    int r = row[e], c = col[e];
    float ex = __expf(edge_score(sA, sB, r, c) - dec_f(m_enc[c]));
    float w = ex / (denom[c] + SM_EPS);
    atomicAdd(&xp[c * DCH + d], x[r * DCH + d] * w);
}

// ---------------- K8: per-node LEConv dots ----------------
__global__ void k_leconv_dots(const float* __restrict__ xp, const float* __restrict__ le1_w,
                              const float* __restrict__ le1_b, const float* __restrict__ le2_w,
                              const float* __restrict__ le3_w, float* __restrict__ a,
                              float* __restrict__ b, float* __restrict__ s3, int n) {
    int node = (blockIdx.x * blockDim.x + threadIdx.x) >> 5;
    int lane = threadIdx.x & 31;
    if (node >= n) return;
    float p0 = xp[node * DCH + lane];
    float p1 = xp[node * DCH + lane + 32];
    float da = wave_sum32(p0 * le1_w[lane] + p1 * le1_w[lane + 32]);
    float db = wave_sum32(p0 * le2_w[lane] + p1 * le2_w[lane + 32]);
    float d3 = wave_sum32(p0 * le3_w[lane] + p1 * le3_w[lane + 32]);
    if (lane == 0) { a[node] = da + le1_b[0]; b[node] = db; s3[node] = d3; }
}

// ---------------- K9: agg = segment_sum(a[row] - b[col], col) ----------------
__global__ void k_edge_agg(const int* __restrict__ row, const int* __restrict__ col,
                           const float* __restrict__ a, const float* __restrict__ b,
                           float* __restrict__ agg, int E) {
    int e = blockIdx.x * blockDim.x + threadIdx.x;
    if (e >= E) return;
    int r = row[e], c = col[e];
    atomicAdd(&agg[c], a[r] - b[c]);
}

// ---------------- K10: fit = sigmoid(agg + s3 + le3_b); out = [xp*fit, fit] ----------------
__global__ void k_finalize(const float* __restrict__ xp, const float* __restrict__ agg,
                           const float* __restrict__ s3, const float* __restrict__ le3_b,
                           float* __restrict__ out, int n) {
    int tid = blockIdx.x * blockDim.x + threadIdx.x;
    int i = tid >> 6, d = tid & 63;
    if (i >= n) return;
    float z = agg[i] + s3[i] + le3_b[0];
    float fit = 1.0f / (1.0f + __expf(-z));
    out[i * DCH + d] = xp[i * DCH + d] * fit;
    if (d == 0) out[n * DCH + i] = fit;
}

extern "C" void kernel_launch(void* const* d_in, const int* in_sizes, int n_in,
                              void* d_out, int out_size, void* d_ws, size_t ws_size,
                              hipStream_t stream) {
    const float* x     = (const float*)d_in[0];
    const int*   ei    = (const int*)d_in[1];
    const float* lin_w = (const float*)d_in[2];
    const float* lin_b = (const float*)d_in[3];
    const float* att_w = (const float*)d_in[4];
    const float* att_b = (const float*)d_in[5];
    const float* le1_w = (const float*)d_in[6];
    const float* le1_b = (const float*)d_in[7];
    const float* le2_w = (const float*)d_in[8];
    const float* le3_w = (const float*)d_in[9];
    const float* le3_b = (const float*)d_in[10];
    float* out = (float*)d_out;

    const int n = in_sizes[0] / DCH;     // 50000
    const int E = in_sizes[1] / 2;       // 850000
    const int* row = ei;                 // edge_index[0] = sources j
    const int* col = ei + E;             // edge_index[1] = targets i

    // workspace layout
    char* ws = (char*)d_ws;
    unsigned* xq_enc = (unsigned*)ws;                                  // n*64 u32
    float*    buf1   = (float*)(ws + (size_t)n * DCH * 4);             // xqlin -> xp (reused)
    float*    sm     = (float*)(ws + (size_t)n * DCH * 8);             // 8 * n floats
    float*    sA    = sm + 0 * (size_t)n;
    float*    sB    = sm + 1 * (size_t)n;
    unsigned* m_enc = (unsigned*)(sm + 2 * (size_t)n);
    float*    denom = sm + 3 * (size_t)n;
    float*    a_    = sm + 4 * (size_t)n;
    float*    b_    = sm + 5 * (size_t)n;
    float*    s3    = sm + 6 * (size_t)n;
    float*    agg   = sm + 7 * (size_t)n;

    const int TB = 256;
    int gN64 = (n * DCH + TB - 1) / TB;
    int gE64 = (E * DCH + TB - 1) / TB;   // 54.4M threads, fits int
    int gE   = (E + TB - 1) / TB;
    int gNwv = (n + 7) / 8;               // one wave32 per node, 8 waves/block
    int tiles = (n + 15) / 16;
    int gGemm = (tiles + 7) / 8;          // 8 waves/block, one 16-row tile per wave

    k_init<<<gN64, TB, 0, stream>>>(xq_enc, m_enc, denom, agg, n);
    k_segmax<<<gE64, TB, 0, stream>>>(x, row, col, xq_enc, E);
    k_gemm_wmma<<<gGemm, TB, 0, stream>>>(xq_enc, lin_w, lin_b, buf1, n);
    k_node_scores<<<gNwv, TB, 0, stream>>>(buf1, x, att_w, att_b, sA, sB, n);
    k_edge_max<<<gE, TB, 0, stream>>>(row, col, sA, sB, m_enc, E);
    k_edge_denom<<<gE, TB, 0, stream>>>(row, col, sA, sB, m_enc, denom, E);
    k_edge_scatter<<<gE64, TB, 0, stream>>>(x, row, col, sA, sB, m_enc, denom, buf1, E);
    k_leconv_dots<<<gNwv, TB, 0, stream>>>(buf1, le1_w, le1_b, le2_w, le3_w, a_, b_, s3, n);
    k_edge_agg<<<gE, TB, 0, stream>>>(row, col, a_, b_, agg, E);
    k_finalize<<<gN64, TB, 0, stream>>>(buf1, agg, s3, le3_b, out, n);
}